// Attn_76708115907054
// MI455X (gfx1250) — compile-verified
//
#include <hip/hip_runtime.h>
#include <hip/hip_bf16.h>
#include <math.h>

#define HIDDEN 512
#define BATCH  64
#define TLEN   4096

typedef __attribute__((ext_vector_type(2))) float v2f;
typedef __attribute__((ext_vector_type(4))) float v4f;
typedef __attribute__((ext_vector_type(8))) float v8f;

// ---------------------------------------------------------------------------
// Stage 1: v[b,h] = sum_o hidden[b,o] * W[o,h]   (M=64, N=512, K=512 fp32 GEMM)
// One wave per 16x16 output tile, K-loop of V_WMMA_F32_16X16X4_F32.
// A 16x4 layout (ISA 7.12.2): VGPR0 = K=k0+2*hi, VGPR1 = K=k0+2*hi+1,
// rows M = lane&15.  B 4x16: row K striped across lanes within a VGPR
// (VGPR0 -> K=k0+2*hi, VGPR1 -> K=k0+2*hi+1, N = n0+(lane&15)).
// D 16x16 f32: VGPR r -> rows m0+r (lanes 0-15) and m0+8+r (lanes 16-31).
// ---------------------------------------------------------------------------
__global__ __launch_bounds__(128) void proj_gemm_wmma(
    const float* __restrict__ hidden, const float* __restrict__ W,
    float* __restrict__ v)
{
    const int wave   = (blockIdx.x * blockDim.x + threadIdx.x) >> 5; // 0..127
    const int lane   = threadIdx.x & 31;
    const int tile_m = wave >> 5;   // 0..3   (64/16)
    const int tile_n = wave & 31;   // 0..31  (512/16)
    const int m0 = tile_m * 16, n0 = tile_n * 16;
    const int lo = lane & 15, hi = lane >> 4;

    v8f acc = {};
    const float* Arow = hidden + (m0 + lo) * HIDDEN;

    for (int k0 = 0; k0 < HIDDEN; k0 += 4) {
        const int ka = k0 + 2 * hi;
        v2f a = *(const v2f*)(Arow + ka);          // contiguous K pair -> b64 load
        v2f b;
        b.x = W[(ka    ) * HIDDEN + n0 + lo];
        b.y = W[(ka + 1) * HIDDEN + n0 + lo];
        acc = __builtin_amdgcn_wmma_f32_16x16x4_f32(
            /*neg_a=*/false, a, /*neg_b=*/false, b,
            /*c_mod=*/(short)0, acc, /*reuse_a=*/false, /*reuse_b=*/false);
    }

#pragma unroll
    for (int r = 0; r < 8; ++r) {
        const int row = m0 + r + 8 * hi;
        v[row * HIDDEN + n0 + lo] = acc[r];
    }
}

// ---------------------------------------------------------------------------
// Stage 2a: c[b] = bias . hidden[b]  (softmax-invariant, kept for exactness)
// ---------------------------------------------------------------------------
__global__ __launch_bounds__(32) void bias_dot(
    const float* __restrict__ hidden, const float* __restrict__ bias,
    float* __restrict__ c)
{
    const int b = blockIdx.x, lane = threadIdx.x;
    float s = 0.f;
#pragma unroll
    for (int k = lane; k < HIDDEN; k += 32)
        s += hidden[b * HIDDEN + k] * bias[k];
#pragma unroll
    for (int off = 16; off; off >>= 1) s += __shfl_xor(s, off, 32);
    if (lane == 0) c[b] = s;
}

// ---------------------------------------------------------------------------
// Stage 2b: scores[b,t] = enc[b,t] . v[b] + c[b]      (bandwidth-bound stage)
// One wave per t: lane L holds v[b, p*128 + 4L .. 4L+3] (p=0..3) in registers;
// each pass reads 512 contiguous bytes per wave via b128 non-temporal loads
// (enc is 512 MB streamed once -> bypass cache retention).
// Block = 8 waves x 16 t  =>  grid (4096/128, 64).
// ---------------------------------------------------------------------------
#define T_PER_WAVE 16
__global__ __launch_bounds__(256) void scores_stream(
    const float* __restrict__ enc, const float* __restrict__ v,
    const float* __restrict__ c, float* __restrict__ scores)
{
    const int b      = blockIdx.y;
    const int wv     = threadIdx.x >> 5;
    const int lane   = threadIdx.x & 31;
    const int t_base = blockIdx.x * (8 * T_PER_WAVE) + wv * T_PER_WAVE;

    v4f vr[4];
    const v4f* vb = (const v4f*)(v + b * HIDDEN);
#pragma unroll
    for (int p = 0; p < 4; ++p) vr[p] = vb[p * 32 + lane];

    const float cb = c[b];
    const float* encb = enc + (size_t)b * TLEN * HIDDEN;

    for (int i = 0; i < T_PER_WAVE; ++i) {
        const int t = t_base + i;
        const v4f* e = (const v4f*)(encb + (size_t)t * HIDDEN);
        float s = 0.f;
#pragma unroll
        for (int p = 0; p < 4; ++p) {
            v4f ev = __builtin_nontemporal_load(&e[p * 32 + lane]);
            s += ev.x * vr[p].x + ev.y * vr[p].y + ev.z * vr[p].z + ev.w * vr[p].w;
        }
#pragma unroll
        for (int off = 16; off; off >>= 1) s += __shfl_xor(s, off, 32);
        if (lane == 0) scores[(size_t)b * TLEN + t] = s + cb;
    }
}

// ---------------------------------------------------------------------------
// Stage 3: in-place row softmax over T=4096 (one block of 256 thr per batch)
// ---------------------------------------------------------------------------
__global__ __launch_bounds__(256) void softmax_rows(float* __restrict__ scores)
{
    __shared__ float red[8];
    const int b    = blockIdx.x;
    const int tid  = threadIdx.x;
    const int lane = tid & 31;
    const int wv   = tid >> 5;
    float* row = scores + (size_t)b * TLEN;

    float vals[16];
    float m = -INFINITY;
#pragma unroll
    for (int i = 0; i < 16; ++i) {
        vals[i] = row[tid + i * 256];
        m = fmaxf(m, vals[i]);
    }
#pragma unroll
    for (int off = 16; off; off >>= 1) m = fmaxf(m, __shfl_xor(m, off, 32));
    if (lane == 0) red[wv] = m;
    __syncthreads();
    float bm = red[0];
#pragma unroll
    for (int w = 1; w < 8; ++w) bm = fmaxf(bm, red[w]);
    __syncthreads();

    float s = 0.f;
#pragma unroll
    for (int i = 0; i < 16; ++i) {
        vals[i] = __expf(vals[i] - bm);
        s += vals[i];
    }
#pragma unroll
    for (int off = 16; off; off >>= 1) s += __shfl_xor(s, off, 32);
    if (lane == 0) red[wv] = s;
    __syncthreads();
    float bs = 0.f;
#pragma unroll
    for (int w = 0; w < 8; ++w) bs += red[w];
    const float inv = 1.f / bs;
#pragma unroll
    for (int i = 0; i < 16; ++i) row[tid + i * 256] = vals[i] * inv;
}

// ---------------------------------------------------------------------------
extern "C" void kernel_launch(void* const* d_in, const int* in_sizes, int n_in,
                              void* d_out, int out_size, void* d_ws, size_t ws_size,
                              hipStream_t stream)
{
    const float* hidden = (const float*)d_in[0];  // [64, 512]
    const float* enc    = (const float*)d_in[1];  // [64, 4096, 512]
    const float* W      = (const float*)d_in[2];  // [512, 512]
    const float* bias   = (const float*)d_in[3];  // [512]
    float* out = (float*)d_out;                   // [64, 1, 4096] == scores, softmaxed in place

    float* v = (float*)d_ws;                      // 64*512 floats = 128 KB
    float* c = v + BATCH * HIDDEN;                // 64 floats

    // 1) v = hidden @ W   (WMMA f32 GEMM, 128 waves)
    proj_gemm_wmma<<<32, 128, 0, stream>>>(hidden, W, v);
    // 2) c[b] = bias . hidden[b]
    bias_dot<<<BATCH, 32, 0, stream>>>(hidden, bias, c);
    // 3) scores -> d_out (streams 512 MB of enc at HBM rate)
    dim3 g2(TLEN / (8 * T_PER_WAVE), BATCH);
    scores_stream<<<g2, 256, 0, stream>>>(enc, v, c, out);
    // 4) softmax in place
    softmax_rows<<<BATCH, 256, 0, stream>>>(out);
}